// RNNDecoder_25546465476706
// MI455X (gfx1250) — compile-verified
//
#include <hip/hip_runtime.h>
#include <math.h>

typedef __attribute__((ext_vector_type(2))) float v2f;
typedef __attribute__((ext_vector_type(8))) float v8f;

#define EMB     512
#define HID     1024
#define VOCAB   32000
#define BATCH   64
#define ENC_LEN 128
#define ENC_DIM 2048
#define G3      (3*HID)          // 3072
#define IN0     (ENC_DIM+EMB)    // 2560
#define FINAL_K (HID+ENC_DIM)    // 3072

// ---------------------------------------------------------------------------
// Attention: betas -> softmax -> context; also emb gather. One block per batch.
// ---------------------------------------------------------------------------
__global__ __launch_bounds__(256)
void attn_kernel(const int* __restrict__ x, const float* __restrict__ h,
                 const float* __restrict__ enc, const float* __restrict__ emb_table,
                 const float* __restrict__ attn_w, const float* __restrict__ attn_b,
                 float* __restrict__ attn_out, float* __restrict__ rnn_in,
                 float* __restrict__ final_in)
{
    const int b   = blockIdx.x;
    const int tid = threadIdx.x;
    __shared__ float sm[256];
    __shared__ float beta[ENC_LEN];
    __shared__ float wgt[ENC_LEN];

    // embedding gather into rnn_in[:, 0:512]
    const int tok = x[b];
    for (int j = tid; j < EMB; j += 256)
        rnn_in[(size_t)b * IN0 + j] = emb_table[(size_t)tok * EMB + j];

    // hdot = dot(h[1][b], w_h) + attn_b
    const float* hrow = h + ((size_t)BATCH + b) * HID;   // layer 1
    float acc = 0.f;
    for (int j = tid; j < HID; j += 256) acc += hrow[j] * attn_w[j];
    sm[tid] = acc; __syncthreads();
    for (int s = 128; s > 0; s >>= 1) { if (tid < s) sm[tid] += sm[tid + s]; __syncthreads(); }
    const float hdot = sm[0] + attn_b[0];
    __syncthreads();

    // betas: each wave handles 16 s-values; coalesced 2048-long dots
    const int wave = tid >> 5, lane = tid & 31;
    const float* we = attn_w + HID;
    const float* eb = enc + (size_t)b * ENC_LEN * ENC_DIM;
    for (int s = wave * 16; s < wave * 16 + 16; ++s) {
        const float* er = eb + (size_t)s * ENC_DIM;
        float a = 0.f;
        for (int e = lane; e < ENC_DIM; e += 32) a += er[e] * we[e];
        #pragma unroll
        for (int off = 16; off > 0; off >>= 1) a += __shfl_xor(a, off, 32);
        if (lane == 0) beta[s] = a + hdot;
    }
    __syncthreads();

    // softmax over 128 (redundant per-thread LDS scan; broadcast-friendly)
    float mx = -INFINITY;
    for (int s = 0; s < ENC_LEN; ++s) mx = fmaxf(mx, beta[s]);
    float sum = 0.f;
    for (int s = 0; s < ENC_LEN; ++s) sum += expf(beta[s] - mx);
    const float inv = 1.f / sum;
    if (tid < ENC_LEN) {
        float w = expf(beta[tid] - mx) * inv;
        wgt[tid] = w;
        attn_out[(size_t)b * ENC_LEN + tid] = w;
    }
    __syncthreads();

    // context[b, e] = sum_s wgt[s] * enc[b, s, e]  (coalesced over e)
    float ctx[ENC_DIM / 256];
    #pragma unroll
    for (int i = 0; i < ENC_DIM / 256; ++i) ctx[i] = 0.f;
    for (int s = 0; s < ENC_LEN; ++s) {
        const float w = wgt[s];
        const float* er = eb + (size_t)s * ENC_DIM;
        #pragma unroll
        for (int i = 0; i < ENC_DIM / 256; ++i) ctx[i] += w * er[tid + 256 * i];
    }
    #pragma unroll
    for (int i = 0; i < ENC_DIM / 256; ++i) {
        const int e = tid + 256 * i;
        rnn_in[(size_t)b * IN0 + EMB + e]        = ctx[i];
        final_in[(size_t)b * FINAL_K + HID + e]  = ctx[i];
    }
}

// ---------------------------------------------------------------------------
// C[64 x N] = A[64 x K] @ W[N x K]^T + bias  via V_WMMA_F32_16X16X4_F32.
// One wave per 16-wide N strip; MT m-tiles of 16 rows each (MT*16 <= 64).
// Fragment layouts per CDNA5 ISA 7.12.2:
//   A 16x4: lanes 0-15 take K={k,k+1}, lanes 16-31 take K={k+2,k+3}
//   B 4x16: mirrored (lane = N, halves split K)
//   C/D:    VGPR j -> M=j (lanes 0-15) / M=8+j (lanes 16-31), N = lane&15
// ---------------------------------------------------------------------------
template<int MT>
__global__ __launch_bounds__(32)
void gemm_wmma(const float* __restrict__ A, int lda,
               const float* __restrict__ W, int ldw,
               const float* __restrict__ bias,
               float* __restrict__ C, int ldc, int K)
{
    const int lane = threadIdx.x;
    const int l    = lane & 15;
    const int half = lane >> 4;
    const int n0   = blockIdx.x * 16;
    const int m0   = blockIdx.y * (MT * 16);

    v8f acc[MT];
    #pragma unroll
    for (int t = 0; t < MT; ++t)
        #pragma unroll
        for (int j = 0; j < 8; ++j) acc[t][j] = 0.f;

    const float* wp = W + (size_t)(n0 + l) * ldw + half * 2;
    const float* ap[MT];
    #pragma unroll
    for (int t = 0; t < MT; ++t)
        ap[t] = A + (size_t)(m0 + t * 16 + l) * lda + half * 2;

    for (int k = 0; k < K; k += 4) {
        const v2f bf = *(const v2f*)wp; wp += 4;
        #pragma unroll
        for (int t = 0; t < MT; ++t) {
            const v2f af = *(const v2f*)ap[t]; ap[t] += 4;
            acc[t] = __builtin_amdgcn_wmma_f32_16x16x4_f32(
                false, af, false, bf, (short)0, acc[t], false, false);
        }
    }

    const float bv = bias ? bias[n0 + l] : 0.f;
    #pragma unroll
    for (int t = 0; t < MT; ++t) {
        #pragma unroll
        for (int j = 0; j < 8; ++j) {
            const int m = m0 + t * 16 + half * 8 + j;
            C[(size_t)m * ldc + n0 + l] = acc[t][j] + bv;
        }
    }
}

// ---------------------------------------------------------------------------
// GRU gate combine: h' = (1-z)*n + z*h_prev
// ---------------------------------------------------------------------------
__global__ __launch_bounds__(256)
void gru_combine(const float* __restrict__ gi, const float* __restrict__ gh,
                 const float* __restrict__ hprev, float* __restrict__ hout,
                 float* __restrict__ extra, int extra_ld)
{
    const int idx = blockIdx.x * 256 + threadIdx.x;
    const int b = idx / HID, j = idx % HID;
    const size_t g = (size_t)b * G3 + j;
    const float ir = gi[g], iz = gi[g + HID], inn = gi[g + 2 * HID];
    const float hr = gh[g], hz = gh[g + HID], hn  = gh[g + 2 * HID];
    const float r = 1.f / (1.f + expf(-(ir + hr)));
    const float z = 1.f / (1.f + expf(-(iz + hz)));
    const float n = tanhf(inn + r * hn);
    const float hp = hprev[(size_t)b * HID + j];
    const float hv = (1.f - z) * n + z * hp;
    hout[(size_t)b * HID + j] = hv;
    if (extra) extra[(size_t)b * extra_ld + j] = hv;
}

// ---------------------------------------------------------------------------
// In-place log_softmax over each 32000-wide row.
// ---------------------------------------------------------------------------
__global__ __launch_bounds__(256)
void logsoftmax_kernel(float* __restrict__ logits)
{
    const int row = blockIdx.x, tid = threadIdx.x;
    float* p = logits + (size_t)row * VOCAB;
    __shared__ float sm[256];

    float mx = -INFINITY;
    for (int j = tid; j < VOCAB; j += 256) mx = fmaxf(mx, p[j]);
    sm[tid] = mx; __syncthreads();
    for (int s = 128; s > 0; s >>= 1) { if (tid < s) sm[tid] = fmaxf(sm[tid], sm[tid + s]); __syncthreads(); }
    mx = sm[0]; __syncthreads();

    float sum = 0.f;
    for (int j = tid; j < VOCAB; j += 256) sum += expf(p[j] - mx);
    sm[tid] = sum; __syncthreads();
    for (int s = 128; s > 0; s >>= 1) { if (tid < s) sm[tid] += sm[tid + s]; __syncthreads(); }
    const float lse = mx + logf(sm[0]);
    __syncthreads();

    for (int j = tid; j < VOCAB; j += 256) p[j] = p[j] - lse;
}

// ---------------------------------------------------------------------------
extern "C" void kernel_launch(void* const* d_in, const int* in_sizes, int n_in,
                              void* d_out, int out_size, void* d_ws, size_t ws_size,
                              hipStream_t stream)
{
    (void)in_sizes; (void)n_in; (void)out_size; (void)ws_size;
    const int*   x    = (const int*)  d_in[0];
    const float* h    = (const float*)d_in[1];
    const float* enc  = (const float*)d_in[2];
    const float* embt = (const float*)d_in[3];
    const float* Wih0 = (const float*)d_in[4];
    const float* Whh0 = (const float*)d_in[5];
    const float* bih0 = (const float*)d_in[6];
    const float* bhh0 = (const float*)d_in[7];
    const float* Wih1 = (const float*)d_in[8];
    const float* Whh1 = (const float*)d_in[9];
    const float* bih1 = (const float*)d_in[10];
    const float* bhh1 = (const float*)d_in[11];
    const float* attw = (const float*)d_in[12];
    const float* attb = (const float*)d_in[13];
    const float* outw = (const float*)d_in[14];
    const float* outb = (const float*)d_in[15];

    float* out  = (float*)d_out;
    float* logp = out;                                    // 64*32000
    float* newh = out + (size_t)BATCH * VOCAB;            // 2*64*1024
    float* atto = newh + (size_t)2 * BATCH * HID;         // 64*128

    float* ws       = (float*)d_ws;
    float* rnn_in   = ws;                                 // 64*2560
    float* final_in = ws + (size_t)BATCH * IN0;           // 64*3072

    // gi/gh scratch lives inside the logits region of d_out; the final GEMM
    // fully overwrites it, so this is safe and deterministic.
    float* gi = logp;
    float* gh = logp + (size_t)BATCH * G3;

    attn_kernel<<<BATCH, 256, 0, stream>>>(x, h, enc, embt, attw, attb,
                                           atto, rnn_in, final_in);

    // Layer 0: gi = rnn_in @ Wih0^T + bih0 ; gh = h0_prev @ Whh0^T + bhh0
    gemm_wmma<1><<<dim3(G3 / 16, 4), 32, 0, stream>>>(rnn_in, IN0, Wih0, IN0, bih0, gi, G3, IN0);
    gemm_wmma<1><<<dim3(G3 / 16, 4), 32, 0, stream>>>(h,      HID, Whh0, HID, bhh0, gh, G3, HID);
    gru_combine<<<BATCH * HID / 256, 256, 0, stream>>>(gi, gh, h, newh, nullptr, 0);

    // Layer 1: gi = h0 @ Wih1^T + bih1 ; gh = h1_prev @ Whh1^T + bhh1
    gemm_wmma<1><<<dim3(G3 / 16, 4), 32, 0, stream>>>(newh, HID, Wih1, HID, bih1, gi, G3, HID);
    gemm_wmma<1><<<dim3(G3 / 16, 4), 32, 0, stream>>>(h + (size_t)BATCH * HID, HID, Whh1, HID, bhh1, gh, G3, HID);
    gru_combine<<<BATCH * HID / 256, 256, 0, stream>>>(gi, gh, h + (size_t)BATCH * HID,
                                                       newh + (size_t)BATCH * HID,
                                                       final_in, FINAL_K);

    // Logits: [h1 | context] @ out_w^T + out_b ; out_w streamed from HBM once
    gemm_wmma<4><<<dim3(VOCAB / 16, 1), 32, 0, stream>>>(final_in, FINAL_K, outw, FINAL_K,
                                                         outb, logp, VOCAB, FINAL_K);

    logsoftmax_kernel<<<BATCH, 256, 0, stream>>>(logp);
}